// ReactionTerm_60138132078971
// MI455X (gfx1250) — compile-verified
//
#include <hip/hip_runtime.h>
#include <cstdint>
#include <cstddef>

// Problem constants (from the reference): B=4096, S=1024, N1=4096, N2=16384.
constexpr int kS       = 1024;
constexpr int kN1      = 4096;
constexpr int kN2      = 16384;
constexpr int kThreads = 256;   // 8 wave32 waves per block

typedef float v4f __attribute__((ext_vector_type(4)));
typedef int   v4i __attribute__((ext_vector_type(4)));

// Address-space-qualified pointer types for the async global->LDS builtin.
// (clang prints AS1 as "__device__" and AS3 as "__shared__" in diagnostics;
// the builtin wants typed v4i pointers, per the round-1 error message.)
typedef __attribute__((address_space(1))) v4i gas_v4i;
typedef __attribute__((address_space(3))) v4i las_v4i;

// One workgroup per batch row. y_in row and y_out accumulator both live in LDS
// (4 KB each). Scatter-adds use ds_add_f32 (LDS float atomics, no-return).
// rate_1 / rate_2 stream through with non-temporal b128 loads.
__global__ __launch_bounds__(kThreads) void ReactionTerm_kernel(
    const float* __restrict__ y_in,
    const float* __restrict__ rate_1,
    const float* __restrict__ rate_2,
    const int*   __restrict__ i1r,
    const int*   __restrict__ i1p,
    const int*   __restrict__ i2r,   // [N2,2] row-major
    const int*   __restrict__ i2p,
    float*       __restrict__ y_out)
{
    __shared__ float ly[kS];     // staged y_in row
    __shared__ float lacc[kS];   // y_out accumulator

    const int b = blockIdx.x;
    const int t = threadIdx.x;

    // Zero the accumulator: 4 floats per thread.
    {
        v4f z = {0.f, 0.f, 0.f, 0.f};
        *(v4f*)&lacc[t * 4] = z;
    }

    // Stage y_in[b, :] into LDS (each lane copies 16 contiguous bytes).
    const float* yrow = y_in + (size_t)b * kS;
#if defined(__gfx1250__) && __has_builtin(__builtin_amdgcn_global_load_async_to_lds_b128) && __has_builtin(__builtin_amdgcn_s_wait_asynccnt)
    {
        // gfx1250 async global->LDS path (ASYNCcnt-tracked DMA-style copy).
        __builtin_amdgcn_global_load_async_to_lds_b128(
            (gas_v4i*)(yrow + t * 4),
            (las_v4i*)(&ly[t * 4]),
            /*offset=*/0, /*cpol=*/0);
        __builtin_amdgcn_s_wait_asynccnt(0);
    }
#else
    *(v4f*)&ly[t * 4] = *(const v4f*)(yrow + t * 4);
#endif
    __syncthreads();

    // ---- First-order reactions: term = ly[i1r[n]] * rate_1[b,n] -> lacc[i1p[n]]
    const float* r1 = rate_1 + (size_t)b * kN1;
    for (int base = t * 4; base < kN1; base += kThreads * 4) {
        v4i ir = *(const v4i*)(i1r + base);          // L2-hot (shared by all rows)
        v4i ip = *(const v4i*)(i1p + base);
        v4f rr = __builtin_nontemporal_load((const v4f*)(r1 + base)); // streamed once
        atomicAdd(&lacc[ip.x], ly[ir.x] * rr.x);
        atomicAdd(&lacc[ip.y], ly[ir.y] * rr.y);
        atomicAdd(&lacc[ip.z], ly[ir.z] * rr.z);
        atomicAdd(&lacc[ip.w], ly[ir.w] * rr.w);
    }

    // ---- Second-order reactions: term = ly[r0]*ly[r1]*rate_2[b,n] -> lacc[i2p[n]]
    const float* r2 = rate_2 + (size_t)b * kN2;
    for (int base = t * 4; base < kN2; base += kThreads * 4) {
        // Prefetch the next streaming tile of rate_2 (global_prefetch_b8).
        if (base + kThreads * 4 < kN2)
            __builtin_prefetch(r2 + base + kThreads * 4, 0, 1);

        // 4 reactions -> 8 consecutive reactant indices (two aligned int4s).
        v4i ira = *(const v4i*)(i2r + 2 * base);     // n0.r0 n0.r1 n1.r0 n1.r1
        v4i irb = *(const v4i*)(i2r + 2 * base + 4); // n2.r0 n2.r1 n3.r0 n3.r1
        v4i ip  = *(const v4i*)(i2p + base);
        v4f rr  = __builtin_nontemporal_load((const v4f*)(r2 + base)); // streamed once
        atomicAdd(&lacc[ip.x], ly[ira.x] * ly[ira.y] * rr.x);
        atomicAdd(&lacc[ip.y], ly[ira.z] * ly[ira.w] * rr.y);
        atomicAdd(&lacc[ip.z], ly[irb.x] * ly[irb.y] * rr.z);
        atomicAdd(&lacc[ip.w], ly[irb.z] * ly[irb.w] * rr.w);
    }

    __syncthreads();  // compiler inserts s_wait_dscnt before the barrier

    // Write the finished row (write-once -> non-temporal store).
    v4f out = *(v4f*)&lacc[t * 4];
    __builtin_nontemporal_store(out, (v4f*)(y_out + (size_t)b * kS + t * 4));
}

extern "C" void kernel_launch(void* const* d_in, const int* in_sizes, int n_in,
                              void* d_out, int out_size, void* d_ws, size_t ws_size,
                              hipStream_t stream)
{
    const float* y_in  = (const float*)d_in[0];
    const float* rate1 = (const float*)d_in[1];
    const float* rate2 = (const float*)d_in[2];
    const int*   i1r   = (const int*)d_in[3];
    const int*   i1p   = (const int*)d_in[4];
    const int*   i2r   = (const int*)d_in[5];
    const int*   i2p   = (const int*)d_in[6];
    float*       y_out = (float*)d_out;

    const int B = in_sizes[0] / kS;   // 4096
    ReactionTerm_kernel<<<B, kThreads, 0, stream>>>(
        y_in, rate1, rate2, i1r, i1p, i2r, i2p, y_out);
}